// DeTree_8461085573696
// MI455X (gfx1250) — compile-verified
//
#include <hip/hip_runtime.h>
#include <hip/hip_bf16.h>

#define F_IN   256
#define NTREES 512
#define DEPTH  6
#define TDIM   3
#define BATCH  512
#define NCOL   (NTREES * DEPTH)   // 3072
#define NLEAF  64

typedef float v2f __attribute__((ext_vector_type(2)));
typedef float v8f __attribute__((ext_vector_type(8)));

// ---------------------------------------------------------------------------
// Kernel 1: exact 1.5-entmax along axis 0 of feat_attention [256, 3072].
// One 256-thread block per column. Bitonic sort + scans in LDS.
// ---------------------------------------------------------------------------
__global__ __launch_bounds__(256) void entmax15_kernel(
    const float* __restrict__ fa, float* __restrict__ cw) {
  const int tid = threadIdx.x;
  const int col = blockIdx.x;

  __shared__ float sh_sorted[F_IN];
  __shared__ float sh_scan[F_IN];
  __shared__ float sh_tau[F_IN];
  __shared__ float sh_red[F_IN];

  float z = fa[tid * NCOL + col] * 0.5f;

  // max-reduce
  sh_red[tid] = z;
  __syncthreads();
  for (int s = F_IN / 2; s > 0; s >>= 1) {
    if (tid < s) sh_red[tid] = fmaxf(sh_red[tid], sh_red[tid + s]);
    __syncthreads();
  }
  z -= sh_red[0];
  __syncthreads();

  // bitonic sort, descending
  sh_sorted[tid] = z;
  __syncthreads();
  for (int k = 2; k <= F_IN; k <<= 1) {
    for (int j = k >> 1; j > 0; j >>= 1) {
      int ixj = tid ^ j;
      if (ixj > tid) {
        float a = sh_sorted[tid];
        float b = sh_sorted[ixj];
        bool swap = ((tid & k) == 0) ? (a < b) : (a > b);  // descending overall
        if (swap) { sh_sorted[tid] = b; sh_sorted[ixj] = a; }
      }
      __syncthreads();
    }
  }
  const float zsv = sh_sorted[tid];

  // inclusive scan of zs
  sh_scan[tid] = zsv;
  __syncthreads();
  for (int off = 1; off < F_IN; off <<= 1) {
    float t = sh_scan[tid];
    if (tid >= off) t += sh_scan[tid - off];
    __syncthreads();
    sh_scan[tid] = t;
    __syncthreads();
  }
  const float s1 = sh_scan[tid];

  // inclusive scan of zs^2
  sh_scan[tid] = zsv * zsv;
  __syncthreads();
  for (int off = 1; off < F_IN; off <<= 1) {
    float t = sh_scan[tid];
    if (tid >= off) t += sh_scan[tid - off];
    __syncthreads();
    sh_scan[tid] = t;
    __syncthreads();
  }
  const float s2 = sh_scan[tid];

  const float rho     = (float)(tid + 1);
  const float mean    = s1 / rho;
  const float mean_sq = s2 / rho;
  const float ss      = rho * (mean_sq - mean * mean);
  const float delta   = (1.0f - ss) / rho;
  const float tau     = mean - sqrtf(fmaxf(delta, 0.0f));
  sh_tau[tid] = tau;

  // support count
  sh_red[tid] = (tau <= zsv) ? 1.0f : 0.0f;
  __syncthreads();
  for (int s = F_IN / 2; s > 0; s >>= 1) {
    if (tid < s) sh_red[tid] += sh_red[tid + s];
    __syncthreads();
  }
  const int kk = (int)(sh_red[0] + 0.5f) - 1;
  const float tau_star = sh_tau[kk];

  float o = fmaxf(z - tau_star, 0.0f);
  cw[tid * NCOL + col] = o * o;
}

// ---------------------------------------------------------------------------
// Kernel 2: feature_values[512,3072] = input[512,256] x choice_weight[256,3072]
// f32 WMMA 16x16x4. One wave owns a 16x64 strip (4 N-tiles sharing A-frag).
// A-frag layout (ISA 7.12.2): lane L -> M = L%16, K = 2*(L/16)+{0,1} in v0/v1.
// B-frag mirrored: lane L -> N = L%16, K rows striped the same way.
// C/D layout: VGPR j, lanes 0-15: (M=j, N=lane); lanes 16-31: (M=j+8, N=lane-16).
// ---------------------------------------------------------------------------
__global__ __launch_bounds__(256) void gemm_f32_wmma_kernel(
    const float* __restrict__ A,   // [BATCH, F_IN]
    const float* __restrict__ W,   // [F_IN, NCOL]
    float* __restrict__ C) {       // [BATCH, NCOL]
  const int lane   = threadIdx.x & 31;
  const int waveId = blockIdx.x * 8 + (threadIdx.x >> 5);  // 0..1535
  const int mTile  = waveId & 31;    // 32 tiles of 16 rows
  const int nStrip = waveId >> 5;    // 48 strips of 64 cols
  const int m0 = mTile * 16;
  const int n0 = nStrip * 64;

  const int l15   = lane & 15;
  const int khalf = (lane >> 4) * 2;   // 0 or 2
  const int mrow  = m0 + l15;

  const v8f vzero = {0.f, 0.f, 0.f, 0.f, 0.f, 0.f, 0.f, 0.f};
  v8f acc[4] = {vzero, vzero, vzero, vzero};

  for (int k = 0; k < F_IN; k += 4) {
    // A fragment: contiguous pair -> one b64 load
    const v2f a = *(const v2f*)(A + mrow * F_IN + k + khalf);
#pragma unroll
    for (int nt = 0; nt < 4; ++nt) {
      const int n = n0 + nt * 16 + l15;
      v2f b;
      b.x = W[(k + khalf) * NCOL + n];
      b.y = W[(k + khalf + 1) * NCOL + n];
      acc[nt] = __builtin_amdgcn_wmma_f32_16x16x4_f32(
          /*neg_a=*/false, a, /*neg_b=*/false, b,
          /*c_mod=*/(short)0, acc[nt],
          /*reuse_a=*/false, /*reuse_b=*/false);
    }
  }

  // store D tiles
  const int mbase = m0 + ((lane >> 4) * 8);
#pragma unroll
  for (int nt = 0; nt < 4; ++nt) {
    const int nn = n0 + nt * 16 + l15;
#pragma unroll
    for (int j = 0; j < 8; ++j) {
      C[(mbase + j) * NCOL + nn] = acc[nt][j];
    }
  }
}

// ---------------------------------------------------------------------------
// Kernel 3: bins -> leaf weights -> response contraction.
// grid (NTREES, BATCH/256), 256 threads; response[t] staged in LDS.
// ---------------------------------------------------------------------------
__global__ __launch_bounds__(256) void eval_kernel(
    const float* __restrict__ fv,    // [BATCH, NCOL]
    const float* __restrict__ th,    // [NTREES, DEPTH]
    const float* __restrict__ lt,    // [NTREES, DEPTH]
    const float* __restrict__ resp,  // [NTREES, TDIM, NLEAF]
    float* __restrict__ out) {       // [BATCH, NTREES*TDIM]
  const int t = blockIdx.x;
  const int b = blockIdx.y * 256 + threadIdx.x;

  __shared__ float sresp[TDIM * NLEAF];
  if (threadIdx.x < TDIM * NLEAF)
    sresp[threadIdx.x] = resp[t * TDIM * NLEAF + threadIdx.x];
  __syncthreads();

  float b0[DEPTH], b1[DEPTH];
#pragma unroll
  for (int d = 0; d < DEPTH; ++d) {
    const float fvv = fv[(size_t)b * NCOL + t * DEPTH + d];
    const float tlv = (fvv - th[t * DEPTH + d]) * expf(-lt[t * DEPTH + d]);
    // bins = clip(0.5*[-tl, tl] + 0.5, -0.5, 1.5); bit==1 picks the -tl slot
    b1[d] = fminf(fmaxf(0.5f * tlv + 0.5f, -0.5f), 1.5f);
    b0[d] = fminf(fmaxf(-0.5f * tlv + 0.5f, -0.5f), 1.5f);
  }

  float a0 = 0.f, a1 = 0.f, a2 = 0.f;
#pragma unroll
  for (int c = 0; c < NLEAF; ++c) {
    float w = (c & 1)  ? b0[0] : b1[0];
    w      *= (c & 2)  ? b0[1] : b1[1];
    w      *= (c & 4)  ? b0[2] : b1[2];
    w      *= (c & 8)  ? b0[3] : b1[3];
    w      *= (c & 16) ? b0[4] : b1[4];
    w      *= (c & 32) ? b0[5] : b1[5];
    a0 += w * sresp[c];
    a1 += w * sresp[NLEAF + c];
    a2 += w * sresp[2 * NLEAF + c];
  }

  float* o = out + (size_t)b * (NTREES * TDIM) + t * TDIM;
  o[0] = a0;
  o[1] = a1;
  o[2] = a2;
}

// ---------------------------------------------------------------------------
extern "C" void kernel_launch(void* const* d_in, const int* in_sizes, int n_in,
                              void* d_out, int out_size, void* d_ws, size_t ws_size,
                              hipStream_t stream) {
  const float* input = (const float*)d_in[0];  // [512, 256]
  const float* fa    = (const float*)d_in[1];  // [256, 3072]
  const float* th    = (const float*)d_in[2];  // [512, 6]
  const float* lt    = (const float*)d_in[3];  // [512, 6]
  const float* resp  = (const float*)d_in[4];  // [512, 3, 64]
  float* out = (float*)d_out;                  // [512, 1536]

  float* cw = (float*)d_ws;                    // [256, 3072] choice weights
  float* fv = cw + (size_t)F_IN * NCOL;        // [512, 3072] feature values

  // 1) entmax per column
  entmax15_kernel<<<NCOL, 256, 0, stream>>>(fa, cw);

  // 2) f32 WMMA GEMM: 1536 waves = 192 blocks x 8 waves
  gemm_f32_wmma_kernel<<<192, 256, 0, stream>>>(input, cw, fv);

  // 3) binning + leaves + response
  eval_kernel<<<dim3(NTREES, BATCH / 256), 256, 0, stream>>>(fv, th, lt, resp, out);
}